// LDW_down_83992380441278
// MI455X (gfx1250) — compile-verified
//
#include <hip/hip_runtime.h>
#include <cstdint>
#include <cstddef>

#ifndef __has_builtin
#define __has_builtin(x) 0
#endif

#define B_   16
#define C_   64
#define H_   257
#define W_   257
#define OH_  128
#define OW_  128
#define OYT  16                    // output rows per block
#define IN_ROWS (2*OYT + 1)        // 33 input rows per block
#define NCHUNK  (OH_/OYT)          // 8 row chunks

#if defined(__HIP_DEVICE_COMPILE__) && __has_builtin(__builtin_amdgcn_tensor_load_to_lds)
#define USE_TDM 1
#else
#define USE_TDM 0
#endif

typedef unsigned int v4u __attribute__((ext_vector_type(4)));
typedef int          v8i __attribute__((ext_vector_type(8)));
typedef int          v4i __attribute__((ext_vector_type(4)));

__global__ __launch_bounds__(128)
void ldw_down_fused(const float* __restrict__ x,
                    const float* __restrict__ low_h, const float* __restrict__ high_h,
                    const float* __restrict__ low_v, const float* __restrict__ high_v,
                    float* __restrict__ out)
{
    // Sole LDS object -> allocated at LDS offset 0 (used by the TDM descriptor).
    __shared__ float tile[IN_ROWS * W_];   // 33*257*4 = 33,924 B

    const int tid   = threadIdx.x;         // 0..127 == ox
    const int chunk = blockIdx.x;          // 0..7
    const int bc    = blockIdx.y;          // 0..1023 == b*C + c
    const int r0    = chunk * (2*OYT);     // first input row of this tile
    const int oy0   = chunk * OYT;

    const float* src = x + (size_t)bc * ((size_t)H_ * W_) + (size_t)r0 * W_;

#if USE_TDM
    // The TDM builtin's LDS writes are invisible to LLVM: with no other store
    // to `tile` in the module, GlobalOpt folds every tile load to undef and
    // deletes the compute (observed: ds=0, valu=7). This opaque, never-taken
    // store defeats that transform and keeps the LDS allocation alive, while
    // costing one scalar compare + untaken branch at runtime.
    {
        int zero = 0;
        asm volatile("" : "+s"(zero));
        if (zero) tile[zero] = 0.0f;
    }
    if (tid == 0) {
        // ---- Tensor DMA descriptor (CDNA5 ISA ch.8) ----
        const uint64_t ga = (uint64_t)(uintptr_t)src;
        v4u g0;
        g0[0] = 1u;                                   // count=1, user mode, no gather
        g0[1] = 0u;                                   // lds_addr = 0 (tile base)
        g0[2] = (unsigned)(ga & 0xFFFFFFFFu);         // global_addr[31:0]
        g0[3] = (unsigned)((ga >> 32) & 0x1FFFFFFu)   // global_addr[56:32]
              | (2u << 30);                           // type = 2 ("image")
        v8i g1;
        g1[0] = 0x00020000;                           // data_size = 2 -> 4 bytes
        g1[1] = (int)((unsigned)(W_ & 0xFFFF) << 16); // tensor_dim0[15:0] = 257
        g1[2] = (int)(0x4000u << 16);                 // dim0[31:16]=0 | tensor_dim1 lo16 (large, no OOB)
        g1[3] = (int)((unsigned)(W_ & 0xFFFF) << 16); // tensor_dim1 hi16=0 | tile_dim0 = 257
        g1[4] = IN_ROWS;                              // tile_dim1 = 33 | tile_dim2 = 0
        g1[5] = W_;                                   // tensor_dim0_stride[31:0] = 257
        g1[6] = 0;                                    // stride0[47:32] | stride1 lo16
        g1[7] = 0;                                    // stride1 hi bits
        v4i g2 = {0, 0, 0, 0};                        // 2D tensor: dims2/3 unused
        v4i g3 = {0, 0, 0, 0};
        v8i g4 = {0, 0, 0, 0, 0, 0, 0, 0};            // clang-23 6-arg form: extra group, zero-filled
        __builtin_amdgcn_tensor_load_to_lds(g0, g1, g2, g3, g4, 0);
        __builtin_amdgcn_s_wait_tensorcnt(0);         // TENSORcnt is per-wave
    }
#else
    // Fallback: cooperative contiguous copy (tile is one flat memory region).
    for (int i = tid; i < IN_ROWS * W_; i += 128)
        tile[i] = src[i];
#endif
    __syncthreads();
    // Compiler-level ordering fence: everything below must really load from LDS.
    asm volatile("" ::: "memory");

    // Filter taps (uniform addresses -> scalar loads).
    const float flh0 = low_h[0],  flh1 = low_h[1];
    const float fhh0 = high_h[0], fhh1 = high_h[1];
    const float flv0 = low_v[0],  flv1 = low_v[1];
    const float fhv0 = high_v[0], fhv1 = high_v[1];

    const int ox = tid;
    const int b  = bc / C_;
    const int c  = bc % C_;
    const size_t plane = (size_t)OH_ * OW_;
    float* o_ll = out + ((size_t)b*4*C_ + 0*C_ + c) * plane;
    float* o_hl = out + ((size_t)b*4*C_ + 1*C_ + c) * plane;
    float* o_lh = out + ((size_t)b*4*C_ + 2*C_ + c) * plane;
    float* o_hh = out + ((size_t)b*4*C_ + 3*C_ + c) * plane;

    const float* t0 = tile + 2*ox;   // per-thread column base in LDS
    // Horizontal lifting of one LDS row (bank-conflict-free: stride 2 across 64 banks).
#define LROW(r) (flh0 * t0[(r)*W_    ] + flh1 * t0[(r)*W_ + 1])
#define HROW(r) (fhh0 * t0[(r)*W_ + 1] + fhh1 * t0[(r)*W_ + 2])

    // Register-carried vertical lifting: each tile row's L/H computed exactly once.
    float pl = LROW(0);
    float ph = HROW(0);
#pragma unroll
    for (int j = 0; j < OYT; ++j) {
        const int ra = 2*j + 1, rb = 2*j + 2;
        const float al = LROW(ra), ah = HROW(ra);
        const float bl = LROW(rb), bh = HROW(rb);
        const size_t o = (size_t)(oy0 + j) * OW_ + ox;   // coalesced across the wave
        // Output is write-once streaming: nontemporal stores keep L2 for the input.
        __builtin_nontemporal_store(flv0*pl + flv1*al, o_ll + o);
        __builtin_nontemporal_store(fhv0*al + fhv1*bl, o_hl + o);
        __builtin_nontemporal_store(flv0*ph + flv1*ah, o_lh + o);
        __builtin_nontemporal_store(fhv0*ah + fhv1*bh, o_hh + o);
        pl = bl; ph = bh;
    }
#undef LROW
#undef HROW
}

extern "C" void kernel_launch(void* const* d_in, const int* in_sizes, int n_in,
                              void* d_out, int out_size, void* d_ws, size_t ws_size,
                              hipStream_t stream) {
    const float* x      = (const float*)d_in[0];
    const float* low_h  = (const float*)d_in[1];
    const float* high_h = (const float*)d_in[2];
    const float* low_v  = (const float*)d_in[3];
    const float* high_v = (const float*)d_in[4];
    float* out = (float*)d_out;

    dim3 grid(NCHUNK, B_ * C_);   // (8, 1024)
    dim3 block(128);
    hipLaunchKernelGGL(ldw_down_fused, grid, block, 0, stream,
                       x, low_h, high_h, low_v, high_v, out);
}